// Subunit_NN_14413910246035
// MI455X (gfx1250) — compile-verified
//
#include <hip/hip_runtime.h>
#include <hip/hip_bf16.h>

typedef __attribute__((ext_vector_type(2))) float v2f;
typedef __attribute__((ext_vector_type(8))) float v8f;
typedef unsigned int v4u __attribute__((ext_vector_type(4)));
typedef int          v8i __attribute__((ext_vector_type(8)));
typedef int          v4i __attribute__((ext_vector_type(4)));

#define T_DATA 50000
#define SUB_NO 20
#define H_NO   16
#define T_NO   200
#define COS_NO 20
#define E_NO   1000
#define I_NO   200
#define PAD    (T_NO - 1)          /* 199 causal zero-pad rows           */
#define SG_COLS 40                 /* interleaved: col 2s = e_s, 2s+1=i_s */
#define SG_ROWS (T_DATA + PAD)
#define NT_TILES (T_DATA / 16)     /* 3125 */
#define WIN (PAD + 16)             /* 215-row signal window per tile      */

// LDS byte-offset of a __shared__ object (generic -> AS(3) -> int)
static __device__ __forceinline__ unsigned lds_offset(const void* p) {
    return (unsigned)(unsigned long long)
        (__attribute__((address_space(3))) const char*)(p);
}

// ---------------------------------------------------------------------------
// conv_kern (already time-flipped, as the reference returns it):
//   kern[o][c][tau] = sum_b W_conv[o][c][b] * basis[b][199-tau]
// ---------------------------------------------------------------------------
__global__ void kern_basis(const float* __restrict__ Wconv,
                           float* __restrict__ out_kern)
{
    int row = blockIdx.x;              // 0..639  == o*2 + c
    int tau = threadIdx.x;
    if (tau >= T_NO) return;
    const float PI = 3.14159265358979323846f;
    float x   = (float)(T_NO - 1 - tau);
    float raw = 5.0f * logf(x + 1.0f);
    float acc = 0.0f;
    for (int b = 0; b < COS_NO; ++b) {
        float phi = 0.5f * PI * (float)b;
        float v = 0.0f;
        if (raw >= phi - PI && raw <= phi + PI)
            v = 0.5f * cosf(raw - phi) + 0.5f;
        acc += Wconv[row * COS_NO + b] * v;
    }
    out_kern[row * T_NO + tau] = acc;
}

// V[t] = root_b + V_o (atomic target, rewritten every launch -> replay-safe);
// zero the causal pad rows of the grouped signal.
__global__ void kern_init(float* __restrict__ V,
                          const float* __restrict__ rootb,
                          const float* __restrict__ Vo,
                          float* __restrict__ sg)
{
    int idx = blockIdx.x * blockDim.x + threadIdx.x;
    if (idx < T_DATA) V[idx] = rootb[0] + Vo[0];
    if (idx < PAD * SG_COLS) sg[idx] = 0.0f;
}

// ---------------------------------------------------------------------------
// Grouping GEMMs: Sg_e = S_e @ C_syn_e^T, Sg_i = S_i @ C_syn_i^T.
// One wave per 16-time tile; V_WMMA_F32_16X16X4_F32; N=20 as 16+4 tiles.
// The 2nd N-tile loads a CLAMPED C row (<=19) unconditionally: WMMA output
// columns are independent, and garbage columns (n>=4) are never stored,
// so no exec-masked loads are needed in the hot K loop.
// ---------------------------------------------------------------------------
__global__ __launch_bounds__(32)
void kern_group(const float* __restrict__ Se, const float* __restrict__ Si,
                const float* __restrict__ Ce, const float* __restrict__ Ci,
                float* __restrict__ sg)
{
    const int lane = threadIdx.x;
    const int n  = lane & 15;
    const int hi = lane >> 4;
    const int t0 = blockIdx.x * 16;
    const int n1 = (n < 4) ? (16 + n) : 19;   // clamped second-tile row

    v8f ae0 = {}, ae1 = {}, ai0 = {}, ai1 = {};

    {   // excitatory: K = 1000
        const float* arow = Se + (size_t)(t0 + n) * E_NO + 2 * hi;
        const float* b0r  = Ce + (size_t)n  * E_NO + 2 * hi;
        const float* b1r  = Ce + (size_t)n1 * E_NO + 2 * hi;
        for (int k0 = 0; k0 < E_NO; k0 += 4) {
            v2f a  = *(const v2f*)(arow + k0);
            v2f b0 = *(const v2f*)(b0r + k0);
            v2f b1 = *(const v2f*)(b1r + k0);
            ae0 = __builtin_amdgcn_wmma_f32_16x16x4_f32(false, a, false, b0,
                                                        (short)0, ae0, false, false);
            ae1 = __builtin_amdgcn_wmma_f32_16x16x4_f32(false, a, false, b1,
                                                        (short)0, ae1, false, false);
        }
    }
    {   // inhibitory: K = 200
        const float* arow = Si + (size_t)(t0 + n) * I_NO + 2 * hi;
        const float* b0r  = Ci + (size_t)n  * I_NO + 2 * hi;
        const float* b1r  = Ci + (size_t)n1 * I_NO + 2 * hi;
        for (int k0 = 0; k0 < I_NO; k0 += 4) {
            v2f a  = *(const v2f*)(arow + k0);
            v2f b0 = *(const v2f*)(b0r + k0);
            v2f b1 = *(const v2f*)(b1r + k0);
            ai0 = __builtin_amdgcn_wmma_f32_16x16x4_f32(false, a, false, b0,
                                                        (short)0, ai0, false, false);
            ai1 = __builtin_amdgcn_wmma_f32_16x16x4_f32(false, a, false, b1,
                                                        (short)0, ai1, false, false);
        }
    }
    // interleaved store: row = [e0,i0,e1,i1,...]; e/i pair as one 8B store
    for (int v = 0; v < 8; ++v) {
        size_t r = (size_t)(PAD + t0 + v + 8 * hi) * SG_COLS;
        v2f p0; p0.x = ae0[v]; p0.y = ai0[v];
        *(v2f*)(sg + r + 2 * n) = p0;
        if (n < 4) {
            v2f p1; p1.x = ae1[v]; p1.y = ai1[v];
            *(v2f*)(sg + r + 32 + 2 * n) = p1;
        }
    }
}

// ---------------------------------------------------------------------------
// Per-(subunit, time-tile): causal conv as GEMM [16h x 400] x [400 x 16t].
// The 215x2 signal window is fetched by the Tensor Data Mover straight into
// LDS (one tensor_load_to_lds per wave, TENSORcnt-tracked), awaited before the
// first kernel-weight loads. Then leaky-ReLU, two 16x16 WMMA MLP layers
// (LDS tile bounce converts C/D layout -> B layout), root dot, atomicAdd V.
// ---------------------------------------------------------------------------
__global__ __launch_bounds__(32)
void kern_conv_mlp(const float* __restrict__ kern,  // flipped [320][2][200]
                   const float* __restrict__ sg,    // [SG_ROWS][40] interleaved
                   const float* __restrict__ ffw,  const float* __restrict__ ffb,
                   const float* __restrict__ ff2w, const float* __restrict__ ff2b,
                   const float* __restrict__ rootw,
                   float* __restrict__ V)
{
    __shared__ float ldsS[WIN][2];        // [row][c] c=0:e, c=1:i (TDM tile)
    __shared__ float ldsX[16][17];        // 16x16 activation tile (padded)

    const int lane = threadIdx.x;
    const int n  = lane & 15;
    const int hi = lane >> 4;
    const int t0 = blockIdx.x * 16;
    const int s  = blockIdx.y;

    // --- TDM: load 2(elem) x 215(rows) tile, row stride 40 elems, 4B data ---
    {
        unsigned long long ga =
            (unsigned long long)(const void*)(sg + (size_t)t0 * SG_COLS + 2 * s);
        v4u g0;
        g0[0] = 1u;                                   // count=1, no gather
        g0[1] = lds_offset(&ldsS[0][0]);              // lds_addr (bytes)
        g0[2] = (unsigned)ga;                         // global_addr[31:0]
        g0[3] = (unsigned)((ga >> 32) & 0x1FFFFFFu)   // global_addr[56:32]
              | (2u << 30);                           // type = 2 ("image")
        v8i g1;
        g1[0] = (int)(2u << 16);                      // data_size=2 (4 bytes)
        g1[1] = (int)(40u << 16);                     // tensor_dim0=40 (lo16)
        g1[2] = (int)(0xFFFFu << 16);                 // tensor_dim1 lo16 (big)
        g1[3] = (int)(0xFu | (2u << 16));             // dim1 hi | tile_dim0=2
        g1[4] = WIN;                                  // tile_dim1=215, dim2=0
        g1[5] = 40;                                   // tensor_dim0_stride=40
        g1[6] = 0;
        g1[7] = 0;
        v4i g2 = {}; v4i g3 = {};                     // unused (<=2D tile)
        v8i g4 = {};                                  // unused trailing group
        __builtin_amdgcn_tensor_load_to_lds(g0, g1, g2, g3, g4, 0);
    }

    // conv_out[t] = sum_tau kern_flipped[tau] * S_pad[t + tau]
    v8f acc = {};
    {
        const float* krow0 = kern + ((size_t)(s * 16 + n) * 2 + 0) * T_NO + 2 * hi;
        const float* krow1 = krow0 + T_NO;
        __builtin_amdgcn_s_wait_tensorcnt(0);         // LDS tile ready
        asm volatile("" ::: "memory");
        for (int d0 = 0; d0 < T_NO; d0 += 4) {
            int rr = n + d0 + 2 * hi;
            v2f a0 = *(const v2f*)(krow0 + d0);
            v2f b0; b0.x = ldsS[rr][0]; b0.y = ldsS[rr + 1][0];
            acc = __builtin_amdgcn_wmma_f32_16x16x4_f32(false, a0, false, b0,
                                                        (short)0, acc, false, false);
            v2f a1 = *(const v2f*)(krow1 + d0);
            v2f b1; b1.x = ldsS[rr][1]; b1.y = ldsS[rr + 1][1];
            acc = __builtin_amdgcn_wmma_f32_16x16x4_f32(false, a1, false, b1,
                                                        (short)0, acc, false, false);
        }
    }
    // leaky relu, stage x[i][t] in LDS
    for (int v = 0; v < 8; ++v) {
        float y = acc[v];
        y = (y > 0.0f) ? y : 0.01f * y;
        ldsX[v + 8 * hi][n] = y;
    }
    __syncthreads();

    // MLP layer 1:  h1 = leaky(ff_w[s] * x + ff_b[s])
    v8f acc1 = {};
    {
        const float* w = ffw + (size_t)s * 256 + n * 16 + 2 * hi;
        for (int k0 = 0; k0 < 16; k0 += 4) {
            v2f a = *(const v2f*)(w + k0);
            v2f b;
            b.x = ldsX[k0 + 2 * hi][n];
            b.y = ldsX[k0 + 2 * hi + 1][n];
            acc1 = __builtin_amdgcn_wmma_f32_16x16x4_f32(false, a, false, b,
                                                         (short)0, acc1, false, false);
        }
    }
    __syncthreads();
    for (int v = 0; v < 8; ++v) {
        float y = acc1[v] + ffb[s * 16 + v + 8 * hi];
        y = (y > 0.0f) ? y : 0.01f * y;
        ldsX[v + 8 * hi][n] = y;
    }
    __syncthreads();

    // MLP layer 2
    v8f acc2 = {};
    {
        const float* w = ff2w + (size_t)s * 256 + n * 16 + 2 * hi;
        for (int k0 = 0; k0 < 16; k0 += 4) {
            v2f a = *(const v2f*)(w + k0);
            v2f b;
            b.x = ldsX[k0 + 2 * hi][n];
            b.y = ldsX[k0 + 2 * hi + 1][n];
            acc2 = __builtin_amdgcn_wmma_f32_16x16x4_f32(false, a, false, b,
                                                         (short)0, acc2, false, false);
        }
    }
    // root partial: p[t] = sum_o h2[o][t] * root_w[s*16+o]
    float p = 0.0f;
    for (int v = 0; v < 8; ++v) {
        float y = acc2[v] + ff2b[s * 16 + v + 8 * hi];
        y = (y > 0.0f) ? y : 0.01f * y;
        p += y * rootw[s * 16 + v + 8 * hi];
    }
    p += __shfl_xor(p, 16, 32);           // combine the two lane halves
    if (hi == 0) atomicAdd(&V[t0 + n], p);
}

// ---------------------------------------------------------------------------
extern "C" void kernel_launch(void* const* d_in, const int* in_sizes, int n_in,
                              void* d_out, int out_size, void* d_ws, size_t ws_size,
                              hipStream_t stream)
{
    (void)in_sizes; (void)n_in; (void)out_size; (void)ws_size;
    const float* Se    = (const float*)d_in[0];
    const float* Si    = (const float*)d_in[1];
    const float* Ce    = (const float*)d_in[2];
    const float* Ci    = (const float*)d_in[3];
    const float* Wconv = (const float*)d_in[4];
    const float* ffw   = (const float*)d_in[5];
    const float* ffb   = (const float*)d_in[6];
    const float* ff2w  = (const float*)d_in[7];
    const float* ff2b  = (const float*)d_in[8];
    const float* rootw = (const float*)d_in[9];
    const float* rootb = (const float*)d_in[10];
    const float* Vo    = (const float*)d_in[11];

    float* out  = (float*)d_out;
    float* V    = out;              // [50000]
    float* kern = out + T_DATA;     // [320*2*200] flipped conv_kern (2nd output)
    float* sg   = (float*)d_ws;     // [SG_ROWS][40] padded grouped signal

    kern_basis<<<dim3(H_NO * SUB_NO * 2), dim3(256), 0, stream>>>(Wconv, kern);
    kern_init <<<dim3((T_DATA + 255) / 256), dim3(256), 0, stream>>>(V, rootb, Vo, sg);
    kern_group<<<dim3(NT_TILES), dim3(32), 0, stream>>>(Se, Si, Ce, Ci, sg);
    kern_conv_mlp<<<dim3(NT_TILES, SUB_NO), dim3(32), 0, stream>>>(
        kern, sg, ffw, ffb, ff2w, ff2b, rootw, V);
}